// GraphNet_56401510531236
// MI455X (gfx1250) — compile-verified
//
#include <hip/hip_runtime.h>
#include <hip/hip_bf16.h>
#include <math.h>

typedef float v2f __attribute__((ext_vector_type(2)));
typedef float v8f __attribute__((ext_vector_type(8)));

#define B_ 4
#define C_ 512
#define N_ 1024
#define K_ 64

// workspace layout (float offsets)
#define WS_INV   0
#define WS_WP1   (WS_INV + K_*C_)
#define WS_WP2   (WS_WP1 + K_*C_)
#define WS_Q     (WS_WP2 + K_*C_)
#define WS_WSUM  (WS_Q + K_)
#define WS_BSQ   (WS_WSUM + B_*K_)
#define WS_OUT2  (WS_BSQ + 16)

// -------------------------------------------------------------------------
// prep: sigma = sigmoid(sig_param); inv = 1/(sigma+eps); inv2 = inv*inv
// pack distance-GEMM weights transposed + pair-interleaved for B-fragments:
//   wp[(c>>1)*K + k] = float2( w[c][k], w[c+1][k] )
// wp1 = -2*anchor*inv2 (multiplies x), wp2 = inv2 (multiplies x^2)
// q[k] = sum_c anchor^2 * inv2.  Also zero wsum / batch-sq accumulators.
// -------------------------------------------------------------------------
__global__ __launch_bounds__(512)
void prep_kernel(const float* __restrict__ anchor,
                 const float* __restrict__ sigp,
                 float* __restrict__ ws) {
  const int k = blockIdx.x;
  const int c = threadIdx.x;            // 512 threads = one per channel
  const float s    = sigp[k * C_ + c];
  const float sig  = 1.0f / (1.0f + __expf(-s));
  const float inv  = 1.0f / (sig + 1e-7f);
  const float inv2 = inv * inv;
  const float a    = anchor[k * C_ + c];

  ws[WS_INV + k * C_ + c] = inv;
  const int pidx = ((c >> 1) * K_ + k) * 2 + (c & 1);
  ws[WS_WP1 + pidx] = -2.0f * a * inv2;
  ws[WS_WP2 + pidx] = inv2;

  __shared__ float red[512];
  red[c] = a * a * inv2;
  __syncthreads();
  for (int off = 256; off > 0; off >>= 1) {
    if (c < off) red[c] += red[c + off];
    __syncthreads();
  }
  if (c == 0) ws[WS_Q + k] = red[0];

  if (k == 0) {                         // zero per-call accumulators
    if (c < B_ * K_) ws[WS_WSUM + c] = 0.0f;
    if (c < 16)      ws[WS_BSQ + c]  = 0.0f;
  }
}

// -------------------------------------------------------------------------
// distance + softmax: each wave computes a 16(n) x 64(k) score tile with
// V_WMMA_F32_16X16X4_F32, accumulating over 2C "features" (x and x^2),
// then softmaxes over k in registers and stores soft_assign[b][k][n].
// Block = 256 threads (8 waves) covers 128 n per block.
// -------------------------------------------------------------------------
__global__ __launch_bounds__(256)
void dist_softmax_kernel(const float* __restrict__ x,
                         float* __restrict__ ws,
                         float* __restrict__ sa) {
  const int b     = blockIdx.x;
  const int ntile = blockIdx.y;                 // 0..7
  const int wv    = threadIdx.x >> 5;           // wave id 0..7
  const int lane  = threadIdx.x & 31;
  const int g     = lane >> 4;                  // half-wave group
  const int m     = lane & 15;
  const int nbase = ntile * 128 + wv * 16;      // 16 n-rows per wave

  const float* xb  = x + (size_t)b * C_ * N_;
  const v2f*   wp1 = (const v2f*)(ws + WS_WP1);
  const v2f*   wp2 = (const v2f*)(ws + WS_WP2);

  v8f acc[4] = {};                              // 4 k-subtiles of 16

  for (int c = 0; c < C_; c += 4) {
    const int c0 = c + 2 * g;                   // A frag: lane-group g holds K=2g,2g+1
    v2f a;
    a.x = xb[(size_t)c0 * N_ + nbase + m];
    a.y = xb[(size_t)(c0 + 1) * N_ + nbase + m];
    v2f a2;
    a2.x = a.x * a.x;
    a2.y = a.y * a.y;
    const int prow = (c >> 1) + g;              // packed float2 row
#pragma unroll
    for (int j = 0; j < 4; ++j) {
      v2f b1 = wp1[prow * K_ + j * 16 + m];
      v2f b2 = wp2[prow * K_ + j * 16 + m];
      acc[j] = __builtin_amdgcn_wmma_f32_16x16x4_f32(
          false, a,  false, b1, (short)0, acc[j], false, false);
      acc[j] = __builtin_amdgcn_wmma_f32_16x16x4_f32(
          false, a2, false, b2, (short)0, acc[j], false, false);
    }
  }

  // scores: -0.5 * (acc + q[k]); D layout: vgpr i, lane -> (n = i+8g, k = 16j+m)
  const float* q = ws + WS_Q;
  float sc[4][8];
#pragma unroll
  for (int j = 0; j < 4; ++j) {
    const float qv = q[j * 16 + m];
#pragma unroll
    for (int i = 0; i < 8; ++i) sc[j][i] = -0.5f * (acc[j][i] + qv);
  }

  // softmax over k (64): 4 in-lane values x 16 lanes of the half-wave
  float sav[4][8];
  float wsum_part[4] = {0.f, 0.f, 0.f, 0.f};
#pragma unroll
  for (int i = 0; i < 8; ++i) {
    float mx = sc[0][i];
#pragma unroll
    for (int j = 1; j < 4; ++j) mx = fmaxf(mx, sc[j][i]);
    for (int msk = 1; msk < 16; msk <<= 1)
      mx = fmaxf(mx, __shfl_xor(mx, msk, 32));
    float e[4];
    float ssum = 0.0f;
#pragma unroll
    for (int j = 0; j < 4; ++j) { e[j] = __expf(sc[j][i] - mx); ssum += e[j]; }
    for (int msk = 1; msk < 16; msk <<= 1)
      ssum += __shfl_xor(ssum, msk, 32);
    const float rs = 1.0f / ssum;
#pragma unroll
    for (int j = 0; j < 4; ++j) {
      const float v = e[j] * rs;
      sav[j][i] = v;
      wsum_part[j] += v;
    }
  }

  // store soft_assign[b][k][n]
  float* sab = sa + (size_t)b * K_ * N_;
#pragma unroll
  for (int j = 0; j < 4; ++j) {
    const int k = j * 16 + m;
#pragma unroll
    for (int i = 0; i < 8; ++i) {
      const int n = nbase + i + 8 * g;
      sab[(size_t)k * N_ + n] = sav[j][i];
    }
  }

  // accumulate wsum[b][k] = sum_n soft_assign
  float* wsum = ws + WS_WSUM + b * K_;
#pragma unroll
  for (int j = 0; j < 4; ++j) {
    float p = wsum_part[j] + __shfl_xor(wsum_part[j], 16, 32);
    if (g == 0) atomicAdd(&wsum[j * 16 + m], p);
  }
}

// -------------------------------------------------------------------------
// aggregate: out2[b][k][c] = sum_n sa[b][k][n] * x[b][c][n]
// WMMA with A = x tile (16c x 4n), B = sa tile (4n x 16k); both operands are
// n-contiguous -> aligned float2 fragment loads.
// -------------------------------------------------------------------------
__global__ __launch_bounds__(256)
void aggregate_kernel(const float* __restrict__ x,
                      const float* __restrict__ sa,
                      float* __restrict__ ws) {
  const int b     = blockIdx.x;
  const int ctile = blockIdx.y;                 // 0..3 (128 c each)
  const int wv    = threadIdx.x >> 5;
  const int lane  = threadIdx.x & 31;
  const int g     = lane >> 4;
  const int m     = lane & 15;
  const int cbase = ctile * 128 + wv * 16;

  const float* xb  = x  + (size_t)b * C_ * N_;
  const float* sab = sa + (size_t)b * K_ * N_;

  v8f acc[4] = {};

  for (int n = 0; n < N_; n += 4) {
    const int n0 = n + 2 * g;
    __builtin_prefetch(xb + (size_t)(cbase + m) * N_ + n0 + 256, 0, 1);
    v2f a = *(const v2f*)(xb + (size_t)(cbase + m) * N_ + n0);
#pragma unroll
    for (int j = 0; j < 4; ++j) {
      v2f bb = *(const v2f*)(sab + (size_t)(j * 16 + m) * N_ + n0);
      acc[j] = __builtin_amdgcn_wmma_f32_16x16x4_f32(
          false, a, false, bb, (short)0, acc[j], false, false);
    }
  }

  float* out2 = ws + WS_OUT2 + (size_t)b * K_ * C_;
#pragma unroll
  for (int j = 0; j < 4; ++j) {
    const int k = j * 16 + m;
#pragma unroll
    for (int i = 0; i < 8; ++i) {
      const int c = cbase + i + 8 * g;
      out2[(size_t)k * C_ + c] = acc[j][i];
    }
  }
}

// -------------------------------------------------------------------------
// finalize: nodes = ((out2 - wsum*anchor)*inv) / (wsum+eps), row-L2-normalize,
// accumulate per-batch squared norm for the flat normalization.
// One block per (b,k); 256 threads, 2 channels each.
// -------------------------------------------------------------------------
__global__ __launch_bounds__(256)
void finalize_kernel(const float* __restrict__ anchor,
                     float* __restrict__ ws,
                     float* __restrict__ nodes_out) {
  const int bk = blockIdx.x;
  const int b  = bk / K_;
  const int k  = bk % K_;
  const int t  = threadIdx.x;

  const float  wsv  = ws[WS_WSUM + b * K_ + k];
  const float* out2 = ws + WS_OUT2 + (size_t)bk * C_;
  const float* inv  = ws + WS_INV + (size_t)k * C_;

  float nd[2];
  float sq = 0.0f;
#pragma unroll
  for (int u = 0; u < 2; ++u) {
    const int c = t * 2 + u;
    const float num = (out2[c] - wsv * anchor[k * C_ + c]) * inv[c];
    const float v   = num / (wsv + 1e-7f);
    nd[u] = v;
    sq += v * v;
  }

  __shared__ float red[256];
  red[t] = sq;
  __syncthreads();
  for (int off = 128; off > 0; off >>= 1) {
    if (t < off) red[t] += red[t + off];
    __syncthreads();
  }
  const float tot = red[0];
  const float rn  = fmaxf(sqrtf(tot), 1e-12f);

  float* o = nodes_out + (size_t)bk * C_;
#pragma unroll
  for (int u = 0; u < 2; ++u) o[t * 2 + u] = nd[u] / rn;

  if (t == 0) atomicAdd(&ws[WS_BSQ + b], tot / (rn * rn));
}

// -------------------------------------------------------------------------
// scale: flat (per-batch) L2 normalization of nodes.
// -------------------------------------------------------------------------
__global__ void scale_kernel(float* __restrict__ nodes_out,
                             const float* __restrict__ ws) {
  const int idx = blockIdx.x * blockDim.x + threadIdx.x;
  if (idx >= B_ * K_ * C_) return;
  const int b = idx / (K_ * C_);
  const float fn = fmaxf(sqrtf(ws[WS_BSQ + b]), 1e-12f);
  nodes_out[idx] /= fn;
}

// -------------------------------------------------------------------------
extern "C" void kernel_launch(void* const* d_in, const int* in_sizes, int n_in,
                              void* d_out, int out_size, void* d_ws, size_t ws_size,
                              hipStream_t stream) {
  const float* x      = (const float*)d_in[0];  // (B,C,H,W)
  const float* anchor = (const float*)d_in[1];  // (K,C)
  const float* sigp   = (const float*)d_in[2];  // (K,C)
  float* out   = (float*)d_out;
  float* nodes = out;                            // B*K*C
  float* sa    = out + (size_t)B_ * K_ * C_;     // B*K*N (soft_assign)
  float* ws    = (float*)d_ws;

  prep_kernel<<<K_, C_, 0, stream>>>(anchor, sigp, ws);
  dist_softmax_kernel<<<dim3(B_, N_ / 128), 256, 0, stream>>>(x, ws, sa);
  aggregate_kernel<<<dim3(B_, C_ / 128), 256, 0, stream>>>(x, sa, ws);
  finalize_kernel<<<B_ * K_, 256, 0, stream>>>(anchor, ws, nodes);
  scale_kernel<<<(B_ * K_ * C_ + 255) / 256, 256, 0, stream>>>(nodes, ws);
}